// Gene_14035953123515
// MI455X (gfx1250) — compile-verified
//
#include <hip/hip_runtime.h>
#include <hip/hip_bf16.h>

typedef __attribute__((ext_vector_type(2))) float v2f;
typedef __attribute__((ext_vector_type(8))) float v8f;

#define HID 128
#define OUTF 64
#define GSTR 192   // G row stride = OUTF + HID
#define NPI 8      // nodes per block in the small-GEMM kernels

// ---------------------------------------------------------------------------
// Scatter-add: out[dst[e], f] += feat[src[e], f]   (f = 0..127)
// Each thread handles one (edge, feature). Destinations (8.2 MB) live in L2.
// ---------------------------------------------------------------------------
__global__ __launch_bounds__(256)
void k_scatter(const float* __restrict__ feat, const int* __restrict__ src,
               const int* __restrict__ dst, float* __restrict__ out, int E) {
  long tid = (long)blockIdx.x * blockDim.x + threadIdx.x;
  long e = tid >> 7;
  int  f = (int)(tid & 127);
  if (e < E) {
    int s = src[e], d = dst[e];
    atomicAdd(&out[(size_t)d * HID + f], feat[(size_t)s * HID + f]);
  }
}

// ---------------------------------------------------------------------------
// h2a = relu(aggx @ w1.T + b1)   (N x 128) @ (128 x 128)
// 128 threads = one output feature each; NPI nodes staged in LDS per pass.
// ---------------------------------------------------------------------------
__global__ __launch_bounds__(128)
void k_mlp1(const float* __restrict__ aggx, const float* __restrict__ w1,
            const float* __restrict__ b1, float* __restrict__ h2a) {
  __shared__ float sIn[NPI][HID];
  int t  = threadIdx.x;
  int n0 = blockIdx.x * NPI;
  for (int r = 0; r < NPI; ++r)
    sIn[r][t] = aggx[(size_t)(n0 + r) * HID + t];
  __syncthreads();

  float acc[NPI];
  float bias = b1[t];
#pragma unroll
  for (int r = 0; r < NPI; ++r) acc[r] = bias;

  const float* wr = w1 + (size_t)t * HID;
  for (int k = 0; k < HID; ++k) {
    float w = wr[k];
#pragma unroll
    for (int r = 0; r < NPI; ++r) acc[r] = fmaf(w, sIn[r][k], acc[r]);
  }
#pragma unroll
  for (int r = 0; r < NPI; ++r)
    h2a[(size_t)(n0 + r) * HID + t] = fmaxf(acc[r], 0.0f);
}

// ---------------------------------------------------------------------------
// Fused: h1 = x@fc_w.T+fc_b ; h2 = aggh@w2.T+b2 ; h = (1-eps)h1 + eps*h2
// Writes G[n] = [ h (64) | x (128) ]  so ret = 0.5 * G @ G.T later.
// 64 threads = one output channel each; NPI nodes staged in LDS per pass.
// ---------------------------------------------------------------------------
__global__ __launch_bounds__(64)
void k_fuse(const float* __restrict__ x, const float* __restrict__ aggh,
            const float* __restrict__ fc_w, const float* __restrict__ fc_b,
            const float* __restrict__ w2, const float* __restrict__ b2,
            const float* __restrict__ epsv, float* __restrict__ G) {
  __shared__ float sX[NPI][HID];
  __shared__ float sA[NPI][HID];
  int t  = threadIdx.x;             // 0..63
  int n0 = blockIdx.x * NPI;
  for (int r = 0; r < NPI; ++r) {
    sX[r][t]      = x[(size_t)(n0 + r) * HID + t];
    sX[r][t + 64] = x[(size_t)(n0 + r) * HID + t + 64];
    sA[r][t]      = aggh[(size_t)(n0 + r) * HID + t];
    sA[r][t + 64] = aggh[(size_t)(n0 + r) * HID + t + 64];
  }
  __syncthreads();

  float a1[NPI], a2[NPI];
  float bb1 = fc_b[t], bb2 = b2[t];
#pragma unroll
  for (int r = 0; r < NPI; ++r) { a1[r] = bb1; a2[r] = bb2; }

  const float* w1r = fc_w + (size_t)t * HID;
  const float* w2r = w2   + (size_t)t * HID;
  for (int k = 0; k < HID; ++k) {
    float wa = w1r[k], wb = w2r[k];
#pragma unroll
    for (int r = 0; r < NPI; ++r) {
      a1[r] = fmaf(wa, sX[r][k], a1[r]);
      a2[r] = fmaf(wb, sA[r][k], a2[r]);
    }
  }
#pragma unroll
  for (int r = 0; r < NPI; ++r) {
    int n = n0 + r;
    float e = epsv[n];
    float h = (1.0f - e) * a1[r] + e * a2[r];
    G[(size_t)n * GSTR + t]        = h;
    G[(size_t)n * GSTR + 64 + t]   = sX[r][t];
    G[(size_t)n * GSTR + 128 + t]  = sX[r][t + 64];
  }
}

// ---------------------------------------------------------------------------
// Column stats of h (first 64 cols of G): deterministic tree reduction.
// One block per channel.  stats[c] = sum, stats[64+c] = sum of squares.
// ---------------------------------------------------------------------------
__global__ __launch_bounds__(256)
void k_stats(const float* __restrict__ G, float* __restrict__ stats, int N) {
  __shared__ float sS[256], sQ[256];
  int c = blockIdx.x;
  int t = threadIdx.x;
  float s = 0.f, q = 0.f;
  for (int i = t; i < N; i += 256) {
    float h = G[(size_t)i * GSTR + c];
    s += h;
    q = fmaf(h, h, q);
  }
  sS[t] = s; sQ[t] = q;
  __syncthreads();
  for (int off = 128; off > 0; off >>= 1) {
    if (t < off) { sS[t] += sS[t + off]; sQ[t] += sQ[t + off]; }
    __syncthreads();
  }
  if (t == 0) { stats[c] = sS[0]; stats[64 + c] = sQ[0]; }
}

// ---------------------------------------------------------------------------
// BatchNorm output: hn = (h - mean) * rsqrt(var + eps) * gamma + beta
// ---------------------------------------------------------------------------
__global__ __launch_bounds__(256)
void k_bn(const float* __restrict__ G, const float* __restrict__ stats,
          const float* __restrict__ gamma, const float* __restrict__ beta,
          float* __restrict__ hn, int N) {
  int idx = blockIdx.x * 256 + threadIdx.x;
  if (idx >= N * OUTF) return;
  int n = idx >> 6, c = idx & 63;
  float inv_n = 1.0f / (float)N;
  float mean  = stats[c] * inv_n;
  float var   = stats[64 + c] * inv_n - mean * mean;
  float inv   = rsqrtf(var + 1e-5f);
  float h     = G[(size_t)n * GSTR + c];
  hn[idx] = (h - mean) * inv * gamma[c] + beta[c];
}

// ---------------------------------------------------------------------------
// ret = 0.5 * G @ G.T  via V_WMMA_F32_16X16X4_F32, one wave per 16x16 tile.
//
// f32 16x16x4 operand layout (wave32):
//   A (16x4):  lanes 0-15 -> rows M=0..15, VGPR0=K0, VGPR1=K1;
//              lanes 16-31 -> same rows,   VGPR0=K2, VGPR1=K3.
//   B (4x16):  mirror: lanes 0-15 -> cols N=0..15 with K0/K1, lanes 16-31 K2/K3.
//   For a Gram matrix both tiles are 16 rows x 4 cols of G with identical
//   per-lane addressing: row = base + (lane&15), col = k + 2*(lane>>4).
//   C/D (16x16 f32, 8 VGPRs): lane<16 -> M=v, N=lane; lane>=16 -> M=8+v, N=lane-16.
//
// G is 12.3 MB -> L2-resident; kernel is bound by the 1.03 GB output store.
// ---------------------------------------------------------------------------
__global__ __launch_bounds__(256)
void k_gram(const float* __restrict__ G, float* __restrict__ ret, int N) {
  int wave = threadIdx.x >> 5;                 // 8 waves: 8 tiles along j
  int lane = threadIdx.x & 31;
  int i0 = blockIdx.y * 16;
  int j0 = (blockIdx.x * 8 + wave) * 16;

  int r  = lane & 15;
  int kh = (lane >> 4) * 2;                    // 0 or 2
  const float* arow = G + (size_t)(i0 + r) * GSTR + kh;
  const float* brow = G + (size_t)(j0 + r) * GSTR + kh;

  v8f c = {};
#pragma unroll
  for (int k = 0; k < GSTR; k += 4) {
    v2f a = *(const v2f*)(arow + k);
    v2f b = *(const v2f*)(brow + k);
    c = __builtin_amdgcn_wmma_f32_16x16x4_f32(
        /*neg_a=*/false, a, /*neg_b=*/false, b,
        /*c_mod=*/(short)0, c, /*reuse_a=*/false, /*reuse_b=*/false);
  }

  int col   = j0 + (lane & 15);
  int rbase = i0 + (lane >> 4) * 8;
#pragma unroll
  for (int v = 0; v < 8; ++v)
    ret[(size_t)(rbase + v) * N + col] = 0.5f * c[v];
}

// ---------------------------------------------------------------------------
extern "C" void kernel_launch(void* const* d_in, const int* in_sizes, int n_in,
                              void* d_out, int out_size, void* d_ws, size_t ws_size,
                              hipStream_t stream) {
  const float* x     = (const float*)d_in[0];
  const int*   src   = (const int*)  d_in[1];
  const int*   dst   = (const int*)  d_in[2];
  const float* fc_w  = (const float*)d_in[3];
  const float* fc_b  = (const float*)d_in[4];
  const float* w1    = (const float*)d_in[5];
  const float* b1    = (const float*)d_in[6];
  const float* w2    = (const float*)d_in[7];
  const float* b2    = (const float*)d_in[8];
  const float* epsv  = (const float*)d_in[9];
  const float* gamma = (const float*)d_in[10];
  const float* beta  = (const float*)d_in[11];

  const int N = in_sizes[0] / HID;    // 16000
  const int E = in_sizes[1];          // 512000

  float* ws    = (float*)d_ws;
  float* aggx  = ws;                              // N*128
  float* h2a   = aggx  + (size_t)N * HID;         // N*128
  float* aggh  = h2a   + (size_t)N * HID;         // N*128
  float* G     = aggh  + (size_t)N * HID;         // N*192
  float* stats = G     + (size_t)N * GSTR;        // 128

  float* ret = (float*)d_out;                     // N*N
  float* hn  = ret + (size_t)N * N;               // N*64

  hipMemsetAsync(aggx, 0, (size_t)N * HID * sizeof(float), stream);
  hipMemsetAsync(aggh, 0, (size_t)N * HID * sizeof(float), stream);

  long scat_threads = (long)E * HID;
  int  scat_blocks  = (int)((scat_threads + 255) / 256);

  k_scatter<<<scat_blocks, 256, 0, stream>>>(x, src, dst, aggx, E);
  k_mlp1   <<<N / NPI, 128, 0, stream>>>(aggx, w1, b1, h2a);
  k_scatter<<<scat_blocks, 256, 0, stream>>>(h2a, src, dst, aggh, E);
  k_fuse   <<<N / NPI, 64, 0, stream>>>(x, aggh, fc_w, fc_b, w2, b2, epsv, G);
  k_stats  <<<OUTF, 256, 0, stream>>>(G, stats, N);
  k_bn     <<<(N * OUTF + 255) / 256, 256, 0, stream>>>(G, stats, gamma, beta, hn, N);

  dim3 grid(N / 128, N / 16);   // 8 tiles per block along j, 1 tile along i
  k_gram   <<<grid, 256, 0, stream>>>(G, ret, N);
}